// HybridModel_79946521248135
// MI455X (gfx1250) — compile-verified
//
#include <hip/hip_runtime.h>
#include <math.h>

typedef float v2f __attribute__((ext_vector_type(2)));
typedef float v8f __attribute__((ext_vector_type(8)));

#define WAVES_PER_BLOCK 8
#define TILE_M 16
#define KDIM 784
#define LDSW_STRIDE 8
// pad 16 extra zero floats so B-fragment reads for N in 8..15 at k=783 stay in-bounds
#define LDSW_ELEMS (KDIM * LDSW_STRIDE + 16)

__launch_bounds__(256)
__global__ void hybrid_fused_kernel(const float* __restrict__ x,
                                    const float* __restrict__ dense_w,
                                    const float* __restrict__ dense_b,
                                    const float* __restrict__ qw,
                                    const float* __restrict__ out_w,
                                    const float* __restrict__ out_b,
                                    float* __restrict__ out)
{
    __shared__ float lds_w[LDSW_ELEMS];                       // 784x6 padded to stride 8, zero tail
    __shared__ float lds_h[WAVES_PER_BLOCK * TILE_M * 8];     // per-wave h tiles

    const int tid  = threadIdx.x;
    const int wave = tid >> 5;
    const int lane = tid & 31;

    // Stage dense_w into LDS (coalesced), zero-pad columns 6..7 and tail.
    for (int i = tid; i < LDSW_ELEMS; i += 256) {
        const int k = i >> 3;
        const int c = i & 7;
        float v = 0.0f;
        if (k < KDIM && c < 6) v = dense_w[k * 6 + c];
        lds_w[i] = v;
    }
    __syncthreads();

    const int tile = blockIdx.x * WAVES_PER_BLOCK + wave;
    const int row0 = tile * TILE_M;

    // f32 WMMA 16x16x4 fragment mapping (wave32):
    //  A (16x4): lanes 0-15 -> M=lane, K={0,1}; lanes 16-31 -> M=lane-16, K={2,3}
    //  B (4x16): lanes 0-15 -> N=lane, K={0,1}; lanes 16-31 -> N=lane-16, K={2,3}
    const int n    = lane & 15;
    const int half = lane >> 4;
    const int koff = half * 2;

    const float* aptr = x + (size_t)(row0 + n) * KDIM + koff;

    v8f acc = {0.f, 0.f, 0.f, 0.f, 0.f, 0.f, 0.f, 0.f};

    #pragma unroll 4
    for (int k = 0; k < KDIM; k += 4) {
        v2f a = *(const v2f*)(aptr + k);          // global_load_b64, 8B aligned
        v2f b;
        b.x = lds_w[(k + koff    ) * LDSW_STRIDE + n];
        b.y = lds_w[(k + koff + 1) * LDSW_STRIDE + n];
        acc = __builtin_amdgcn_wmma_f32_16x16x4_f32(false, a, false, b,
                                                    (short)0, acc, false, false);
    }

    // Bias: every C row in this lane has column N=n.
    float bias = 0.0f;
    if (n < 6) bias = dense_b[n];
    #pragma unroll
    for (int i = 0; i < 8; ++i) acc[i] += bias;

    // Spill h tile (only the 6 real columns) to this wave's LDS region.
    float* hbase = lds_h + wave * (TILE_M * 8);
    if (n < 6) {
        #pragma unroll
        for (int i = 0; i < 8; ++i) {
            const int M = i + half * 8;           // C layout: VGPR i -> M=i (lo lanes), M=i+8 (hi lanes)
            hbase[M * 8 + n] = acc[i];
        }
    }
    __syncthreads();

    // Epilogue: one row per lane (lanes 0..15).
    if (lane < TILE_M) {
        float a[6];
        #pragma unroll
        for (int j = 0; j < 6; ++j) a[j] = hbase[lane * 8 + j];

        float s = 0.0f;
        #pragma unroll
        for (int j = 0; j < 6; ++j) s += a[j] * a[j];
        const float inv = rsqrtf(s);              // v_rsq_f32 (avoid precise-sqrt expansion)
        #pragma unroll
        for (int j = 0; j < 6; ++j) a[j] *= inv;

        const int PI[15] = {0, 1, 0, 2, 1, 3, 0, 2, 4, 1, 3, 0, 2, 1, 0};
        const int PJ[15] = {1, 2, 1, 3, 2, 4, 1, 3, 5, 2, 4, 1, 3, 2, 1};
        #pragma unroll
        for (int kq = 0; kq < 15; ++kq) {
            const float w  = qw[kq];
            const float cw = __cosf(w);           // v_cos_f32: args are ~N(0,0.1^2), no Payne-Hanek needed
            const float sw = __sinf(w);           // v_sin_f32
            const float ai = a[PI[kq]];
            const float aj = a[PJ[kq]];
            a[PI[kq]] =  cw * ai + sw * aj;
            a[PJ[kq]] = -sw * ai + cw * aj;
        }

        float o = out_b[0];
        #pragma unroll
        for (int j = 0; j < 6; ++j) {
            float z = 1.0f - 2.0f * a[j] * a[j];
            if (z != z) z = 0.0f;                 // NaN -> 0, matches reference
            o += z * out_w[j];
        }
        out[row0 + lane] = 1.0f / (1.0f + __expf(-o));
    }
}

extern "C" void kernel_launch(void* const* d_in, const int* in_sizes, int n_in,
                              void* d_out, int out_size, void* d_ws, size_t ws_size,
                              hipStream_t stream) {
    const float* x       = (const float*)d_in[0];   // (65536, 28, 28) fp32
    const float* dense_w = (const float*)d_in[1];   // (784, 6)
    const float* dense_b = (const float*)d_in[2];   // (6,)
    const float* qw      = (const float*)d_in[3];   // (15,)
    const float* out_w   = (const float*)d_in[4];   // (6, 1)
    const float* out_b   = (const float*)d_in[5];   // (1,)
    float* out = (float*)d_out;                     // (65536, 1) fp32

    const int B      = in_sizes[0] / KDIM;          // 65536 rows
    const int tiles  = (B + TILE_M - 1) / TILE_M;   // 4096
    const int blocks = (tiles + WAVES_PER_BLOCK - 1) / WAVES_PER_BLOCK;  // 512

    hybrid_fused_kernel<<<blocks, 256, 0, stream>>>(x, dense_w, dense_b, qw,
                                                    out_w, out_b, out);
}